// TensorSRHT_50362786513438
// MI455X (gfx1250) — compile-verified
//
#include <hip/hip_runtime.h>
#include <hip/hip_bf16.h>

// TensorSRHT for MI455X (gfx1250):
//   out[n,f] = ( prod_d FWHT(x[n]*rad[d])[perm[d][f]] ) / sqrt(8192)
//
// FWHT_4096 = H16 (x) H16 (x) H16, each 16-point Hadamard stage done with
// chained V_WMMA_F32_16X16X4_F32 (D = A x H16, H16 symmetric ±1, generated
// in-register as sign-bit of 1.0f from popc(k&n)). The whole transform runs
// IN PLACE in a 16KB LDS buffer per degree; tile ownership is wave-consistent
// so only one workgroup barrier per degree is needed.
//
// Bandwidth-bound: ~384 MB of unavoidable traffic -> ~16.5 us at 23.3 TB/s.
// Output stores are non-temporal so the 256 MB stream does not evict the
// L2-resident x (128 MB) / rad / perm tables (L2 = 192 MB).

typedef __attribute__((ext_vector_type(2))) float v2f;
typedef __attribute__((ext_vector_type(4))) float v4f;
typedef __attribute__((ext_vector_type(8))) float v8f;
typedef __attribute__((ext_vector_type(4))) int   v4i;

#define N_ROWS 8192
#define D_IN   4096
#define D_F    8192
#define NDEG   3

// H16[k][n] = (-1)^popc(k&n) as a float, built by OR-ing parity into the
// sign bit of 1.0f (no compares / cndmask).
__device__ __forceinline__ float hsign(int k, int n) {
    unsigned parity = (unsigned)(__popc(k & n) & 1);
    return __int_as_float((int)(0x3f800000u | (parity << 31)));
}

__global__ __launch_bounds__(256) void
srht_wmma_kernel(const float* __restrict__ x,
                 const float* __restrict__ rad,
                 const int*   __restrict__ perm,
                 float*       __restrict__ out)
{
    __shared__ float Z[NDEG][D_IN];          // 48 KB LDS

    const int tid  = threadIdx.x;
    const int lane = tid & 31;               // wave32
    const int wave = tid >> 5;               // 8 waves
    const int row  = blockIdx.x;

    const int half = lane >> 4;              // WMMA operand lane-half split
    const int mn   = lane & 15;              // M for A / N for B,D

    // ---- x row chunk cached in registers: 16 floats per thread -------------
    const v4f* xrow = reinterpret_cast<const v4f*>(x + (size_t)row * D_IN);
    v4f xv[4];
#pragma unroll
    for (int e = 0; e < 4; ++e) xv[e] = xrow[tid * 4 + e];

    // ---- B operand = H16 signs per K-slice t (K = 4t + {0,2}/{1,3}) --------
    v2f hb[4];
#pragma unroll
    for (int t = 0; t < 4; ++t) {
        const int k0 = 4 * t + (half ? 2 : 0);
        const int k1 = 4 * t + (half ? 3 : 1);
        hb[t].x = hsign(k0, mn);
        hb[t].y = hsign(k1, mn);
    }

    for (int d = 0; d < NDEG; ++d) {
        float* Zd = Z[d];

        // -- multiply pass: Zd = x * rad[d]  (GAMMA = 1) ---------------------
        // wave w writes exactly blocks 2w,2w+1 that it consumes below:
        // wave-local, LDS in-order per wave -> no barrier.
        {
            const v4f* rrow = reinterpret_cast<const v4f*>(rad + d * D_IN);
            v4f* zv = reinterpret_cast<v4f*>(Zd);
#pragma unroll
            for (int e = 0; e < 4; ++e)
                zv[tid * 4 + e] = xv[e] * rrow[tid * 4 + e];
        }

        // -- modes 0 and 1 (low / mid 4 index bits), fused, in place ---------
#pragma unroll
        for (int ti = 0; ti < 2; ++ti) {
            const int r = wave * 2 + ti;     // 256-element block owned by wave

            // mode 0: D = V x H within block r
            v8f c0 = {};
#pragma unroll
            for (int t = 0; t < 4; ++t) {
                const int k0 = 4 * t + (half ? 2 : 0);
                const int k1 = 4 * t + (half ? 3 : 1);
                v2f a;
                a.x = Zd[256 * r + 16 * mn + k0];
                a.y = Zd[256 * r + 16 * mn + k1];
                c0 = __builtin_amdgcn_wmma_f32_16x16x4_f32(
                        false, a, false, hb[t], (short)0, c0, false, false);
            }
#pragma unroll
            for (int q = 0; q < 8; ++q) {
                const int M = q + (half ? 8 : 0);
                Zd[256 * r + 16 * M + mn] = c0[q];     // D[M][N]
            }

            // mode 1: same block, transposed read + transposed write
            v8f c1 = {};
#pragma unroll
            for (int t = 0; t < 4; ++t) {
                const int k0 = 4 * t + (half ? 2 : 0);
                const int k1 = 4 * t + (half ? 3 : 1);
                v2f a;
                a.x = Zd[256 * r + 16 * k0 + mn];
                a.y = Zd[256 * r + 16 * k1 + mn];
                c1 = __builtin_amdgcn_wmma_f32_16x16x4_f32(
                        false, a, false, hb[t], (short)0, c1, false, false);
            }
#pragma unroll
            for (int q = 0; q < 8; ++q) {
                const int M = q + (half ? 8 : 0);
                Zd[256 * r + 16 * mn + M] = c1[q];     // transposed store
            }
        }

        __syncthreads();   // mode 2 reads all blocks -> cross-wave dependency

        // -- mode 2 (high 4 index bits): 16-column stripe per tile, in place -
#pragma unroll
        for (int ti = 0; ti < 2; ++ti) {
            const int r = wave * 2 + ti;
            v8f c2 = {};
#pragma unroll
            for (int t = 0; t < 4; ++t) {
                const int k0 = 4 * t + (half ? 2 : 0);
                const int k1 = 4 * t + (half ? 3 : 1);
                v2f a;
                a.x = Zd[256 * k0 + 16 * r + mn];
                a.y = Zd[256 * k1 + 16 * r + mn];
                c2 = __builtin_amdgcn_wmma_f32_16x16x4_f32(
                        false, a, false, hb[t], (short)0, c2, false, false);
            }
#pragma unroll
            for (int q = 0; q < 8; ++q) {
                const int M = q + (half ? 8 : 0);
                Zd[256 * mn + 16 * r + M] = c2[q];     // z[a=N][m=16r+M]
            }
        }
        // next degree uses a different LDS buffer -> no barrier needed here
    }

    __syncthreads();       // all three z_d complete

    // ---- gather + product + scale: int4 perm loads, NT float4 stores -------
    const float inv_sqrt_df = 0.011048543456039806f;   // 1/sqrt(8192)
    const v4i* p0v = reinterpret_cast<const v4i*>(perm);
    const v4i* p1v = reinterpret_cast<const v4i*>(perm + D_F);
    const v4i* p2v = reinterpret_cast<const v4i*>(perm + 2 * D_F);
    v4f* orow = reinterpret_cast<v4f*>(out + (size_t)row * D_F);

    for (int f4 = tid; f4 < D_F / 4; f4 += 256) {
        const v4i p0 = p0v[f4];
        const v4i p1 = p1v[f4];
        const v4i p2 = p2v[f4];
        v4f r;
        r.x = Z[0][p0.x] * Z[1][p1.x] * Z[2][p2.x] * inv_sqrt_df;
        r.y = Z[0][p0.y] * Z[1][p1.y] * Z[2][p2.y] * inv_sqrt_df;
        r.z = Z[0][p0.z] * Z[1][p1.z] * Z[2][p2.z] * inv_sqrt_df;
        r.w = Z[0][p0.w] * Z[1][p1.w] * Z[2][p2.w] * inv_sqrt_df;
        __builtin_nontemporal_store(r, &orow[f4]);
    }
}

extern "C" void kernel_launch(void* const* d_in, const int* in_sizes, int n_in,
                              void* d_out, int out_size, void* d_ws, size_t ws_size,
                              hipStream_t stream)
{
    const float* x    = (const float*)d_in[0];   // (8192, 4096) f32
    const float* rad  = (const float*)d_in[1];   // (3, 4096)    f32 (+-1)
    const int*   perm = (const int*)  d_in[2];   // (3, 8192)    i32
    float*       out  = (float*)d_out;           // (8192, 8192) f32

    srht_wmma_kernel<<<dim3(N_ROWS), dim3(256), 0, stream>>>(x, rad, perm, out);
}